// NMS_52372831207837
// MI455X (gfx1250) — compile-verified
//
#include <hip/hip_runtime.h>

// Problem constants (from reference)
#define NBATCH   16
#define NDET     25200
#define NCH      85        // 4 box + 1 obj + 80 classes
#define NCLS     80
#define TOPK     1000
#define TPAD     1008      // TOPK padded to multiple of 16 (63 wmma tiles)
#define NTILE    63
#define SORTM    32768     // NDET padded to power of two
#define CONF_THR 0.3f
#define IOU_THR  0.6f
#define MAX_WH   4096.0f
#define ORD_NEGINF 0x007FFFFFu   // ordf(-inf)

typedef __attribute__((ext_vector_type(2))) float v2f;
typedef __attribute__((ext_vector_type(8))) float v8f;

// monotonic float -> uint transform (descending float == descending uint)
__device__ __forceinline__ unsigned int ordf(float f) {
  unsigned int u = __float_as_uint(f);
  return u ^ ((u >> 31) ? 0xFFFFFFFFu : 0x80000000u);
}
__device__ __forceinline__ float unordf(unsigned int o) {
  unsigned int u = (o & 0x80000000u) ? (o ^ 0x80000000u) : ~o;
  return __uint_as_float(u);
}

// ---------------------------------------------------------------------------
// Kernel 1: decode. One wave32 per detection row (coalesced 85-float read).
// Emits: orderable score key (-inf if below thresholds), class id, xyxy box.
// HBM-bound stage: 137MB in @ 23.3TB/s ~ 6us floor.
// ---------------------------------------------------------------------------
__global__ void nms_decode(const float* __restrict__ x,
                           unsigned int* __restrict__ keys,
                           int* __restrict__ clsout,
                           float4* __restrict__ boxout,
                           int total) {
  int r = blockIdx.x * 8 + (threadIdx.x >> 5);   // 8 waves / 256-thread block
  if (r >= total) return;                         // wave-uniform exit
  int lane = threadIdx.x & 31;
  const float* p = x + (size_t)r * NCH;

  // prefetch one row-block (8 rows) ahead -> global_prefetch_b8
  if (r + 8 < total) __builtin_prefetch(p + (size_t)8 * NCH + lane * 4, 0, 1);

  float v0 = p[lane];                             // elems 0..31
  float v1 = p[lane + 32];                        // elems 32..63
  float v2 = (lane < NCH - 64) ? p[lane + 64] : 0.0f; // elems 64..84

  float obj = __shfl(v0, 4);

  // per-lane argmax over its classes (candidates visited in ascending cls)
  float bestv = -__builtin_inff();
  int   bestc = 1 << 30;
  if (lane >= 5) { float s = v0 * obj; if (s > bestv) { bestv = s; bestc = lane - 5; } }
  { float s = v1 * obj; if (s > bestv) { bestv = s; bestc = lane + 27; } }
  if (lane < NCH - 64) { float s = v2 * obj; if (s > bestv) { bestv = s; bestc = lane + 59; } }

  // wave32 butterfly argmax, ties -> smaller class id (matches jnp.argmax)
  for (int off = 16; off; off >>= 1) {
    float ov = __shfl_xor(bestv, off);
    int   oc = __shfl_xor(bestc, off);
    if (ov > bestv || (ov == bestv && oc < bestc)) { bestv = ov; bestc = oc; }
  }

  float cx = __shfl(v0, 0), cy = __shfl(v0, 1);
  float ww = __shfl(v0, 2), hh = __shfl(v0, 3);

  if (lane == 0) {
    bool valid = (obj > CONF_THR) && (bestv > CONF_THR);
    float score = valid ? bestv : -__builtin_inff();
    keys[r]   = ordf(score);
    clsout[r] = bestc;
    float4 bx;
    bx.x = cx - ww * 0.5f; bx.y = cy - hh * 0.5f;
    bx.z = cx + ww * 0.5f; bx.w = cy + hh * 0.5f;
    boxout[r] = bx;
  }
}

// ---------------------------------------------------------------------------
// Kernel 2: exact top-1000 per batch via full bitonic sort held entirely in
// CDNA5's 320KB LDS (32768 x (u32 key + u16 idx) = 192KB). Descending by
// score, ties broken by ascending index (== lax.top_k semantics).
// ---------------------------------------------------------------------------
__global__ void __launch_bounds__(1024)
nms_topk(const unsigned int* __restrict__ keys,
         const int* __restrict__ cls,
         const float4* __restrict__ box,
         float4* __restrict__ topbox,
         float* __restrict__ topscore,
         int* __restrict__ topcls,
         int* __restrict__ topvalid) {
  extern __shared__ char smem[];
  unsigned int*   skey = (unsigned int*)smem;                 // SORTM u32
  unsigned short* sidx = (unsigned short*)(smem + SORTM * 4); // SORTM u16
  int b = blockIdx.x;

  for (int t = threadIdx.x; t < SORTM; t += blockDim.x) {
    if (t < NDET) { skey[t] = keys[(size_t)b * NDET + t]; sidx[t] = (unsigned short)t; }
    else          { skey[t] = 0u;                          sidx[t] = 0xFFFFu; }
  }
  __syncthreads();

  for (int k = 2; k <= SORTM; k <<= 1) {
    for (int j = k >> 1; j > 0; j >>= 1) {
      for (int i = threadIdx.x; i < SORTM; i += blockDim.x) {
        int ixj = i ^ j;
        if (ixj > i) {
          unsigned int   ka = skey[i], kb = skey[ixj];
          unsigned short ia = sidx[i], ib = sidx[ixj];
          bool aGTb = (ka > kb) || (ka == kb && ia < ib);
          bool bGTa = (kb > ka) || (kb == ka && ib < ia);
          bool sw = ((i & k) == 0) ? bGTa : aGTb;  // (i&k)==0 -> descending run
          if (sw) { skey[i] = kb; skey[ixj] = ka; sidx[i] = ib; sidx[ixj] = ia; }
        }
      }
      __syncthreads();
    }
  }

  // NDET >= TOPK real entries outrank padding, so slots 0..999 are real.
  for (int t = threadIdx.x; t < TOPK; t += blockDim.x) {
    unsigned int key = skey[t];
    int idx = (int)sidx[t];
    size_t src = (size_t)b * NDET + idx;
    size_t dst = (size_t)b * TOPK + t;
    topbox[dst]   = box[src];
    topscore[dst] = unordf(key);
    topcls[dst]   = cls[src];
    topvalid[dst] = (key != ORD_NEGINF) ? 1 : 0;
  }
}

// ---------------------------------------------------------------------------
// Kernel 3: 1000x1000 IoU suppression matrix as LDS bitmask, then greedy
// serial suppression by one wave. Pairwise (area_i + area_j) term computed
// with v_wmma_f32_16x16x4_f32 as a rank-2 outer product per 16x16 tile:
//   A(16x4) row m = [area_row[m], 1, 0, 0],  B(4x16) = [ones; area_col; 0, 0]
//   => D[m][n] = area_row[m] + area_col[n]
// IoU test is division-free: inter > thr*(areaSum - inter + eps), valid since
// the denominator is strictly positive.
// ---------------------------------------------------------------------------
__global__ void __launch_bounds__(1024)
nms_suppress(const float4* __restrict__ topbox,
             const float* __restrict__ topscore,
             const int* __restrict__ topcls,
             const int* __restrict__ topvalid,
             float* __restrict__ out) {
  extern __shared__ char smem[];
  float4*       ob      = (float4*)smem;                                // TPAD boxes (class-offset)
  unsigned int* sup     = (unsigned int*)(smem + TPAD * 16);            // TPAD x 32 words
  unsigned int* keeplds = (unsigned int*)(smem + TPAD * 16 + TPAD * 32 * 4); // 32 words
  int b = blockIdx.x;

  for (int t = threadIdx.x; t < TPAD; t += blockDim.x) {
    float4 v = {0.0f, 0.0f, 0.0f, 0.0f};
    if (t < TOPK) {
      v = topbox[(size_t)b * TOPK + t];
      float off = (float)topcls[(size_t)b * TOPK + t] * MAX_WH;
      v.x += off; v.y += off; v.z += off; v.w += off;
    }
    ob[t] = v;
  }
  for (int t = threadIdx.x; t < TPAD * 32; t += blockDim.x) sup[t] = 0u;
  __syncthreads();

  int wave = threadIdx.x >> 5;
  int lane = threadIdx.x & 31;
  int m16  = lane & 15;

  for (int t = wave; t < NTILE * NTILE; t += (int)(blockDim.x >> 5)) {
    int ti = t / NTILE, tj = t % NTILE;
    if (tj < ti) continue;   // lower triangle: gj > gi never holds (wave-uniform)

    // Build WMMA fragments (A 16x4: V0=K0|K2, V1=K1|K3; B 4x16 analogous).
    float ar = 0.0f, ac = 0.0f;
    if (lane < 16) {
      float4 rb = ob[ti * 16 + m16];
      ar = (rb.z - rb.x) * (rb.w - rb.y);
      float4 cb = ob[tj * 16 + m16];
      ac = (cb.z - cb.x) * (cb.w - cb.y);
    }
    v2f a, bm;
    a.x  = (lane < 16) ? ar   : 0.0f;   // K=0 : area_row[m]   (K=2 zero)
    a.y  = (lane < 16) ? 1.0f : 0.0f;   // K=1 : ones          (K=3 zero)
    bm.x = (lane < 16) ? 1.0f : 0.0f;   // K=0 : ones          (K=2 zero)
    bm.y = (lane < 16) ? ac   : 0.0f;   // K=1 : area_col[n]   (K=3 zero)
    v8f c = {};
    v8f d = __builtin_amdgcn_wmma_f32_16x16x4_f32(false, a, false, bm,
                                                  (short)0, c, false, false);

    int mbase = ti * 16 + ((lane >= 16) ? 8 : 0);
    int gj = tj * 16 + m16;
    float4 cbox = ob[gj];
    for (int r = 0; r < 8; ++r) {
      int gi = mbase + r;
      float4 rbox = ob[gi];
      float ltx = fmaxf(rbox.x, cbox.x), lty = fmaxf(rbox.y, cbox.y);
      float rbx = fminf(rbox.z, cbox.z), rby = fminf(rbox.w, cbox.w);
      float iw = fmaxf(rbx - ltx, 0.0f), ih = fmaxf(rby - lty, 0.0f);
      float inter = iw * ih;
      // iou > thr  <=>  inter > thr * (areaSum - inter + eps)   (denom > 0)
      bool pred = (inter > IOU_THR * (d[r] - inter + 1e-9f)) &&
                  (gj > gi) && (gi < TOPK) && (gj < TOPK);
      unsigned long long bal = __ballot(pred);
      if (lane == 0) {
        unsigned int w32 = (unsigned int)bal;
        int cw = tj >> 1, sh = (tj & 1) * 16;
        int rlo = ti * 16 + r;
        atomicOr(&sup[rlo * 32 + cw],       (w32 & 0xFFFFu) << sh); // rows m=r
        atomicOr(&sup[(rlo + 8) * 32 + cw], (w32 >> 16)     << sh); // rows m=r+8
      }
    }
  }
  __syncthreads();

  // Greedy serial suppression: wave 0 holds the 1000-bit keep mask,
  // lane w owns word w (bits w*32 .. w*32+31).
  if (threadIdx.x < 32) {
    unsigned int kw = 0u;
    for (int k2 = 0; k2 < 32; ++k2) {
      int j = lane * 32 + k2;
      if (j < TOPK && topvalid[(size_t)b * TOPK + j]) kw |= (1u << k2);
    }
    for (int i = 0; i < TOPK; ++i) {
      unsigned int ow = __shfl(kw, i >> 5);
      if ((ow >> (i & 31)) & 1u) kw &= ~sup[i * 32 + lane];
    }
    keeplds[lane] = kw;
  }
  __syncthreads();

  for (int j = threadIdx.x; j < TOPK; j += blockDim.x) {
    unsigned int kw = keeplds[j >> 5];
    float km = (float)((kw >> (j & 31)) & 1u);
    size_t src = (size_t)b * TOPK + j;
    float4 bx = topbox[src];
    float sc = (km != 0.0f) ? topscore[src] : 0.0f;
    float cf = (float)topcls[src] * km;
    float* o = out + src * 6;
    o[0] = bx.x * km; o[1] = bx.y * km; o[2] = bx.z * km; o[3] = bx.w * km;
    o[4] = sc;        o[5] = cf;
  }
}

// ---------------------------------------------------------------------------
extern "C" void kernel_launch(void* const* d_in, const int* in_sizes, int n_in,
                              void* d_out, int out_size, void* d_ws, size_t ws_size,
                              hipStream_t stream) {
  (void)in_sizes; (void)n_in; (void)out_size; (void)ws_size;
  const float* x = (const float*)d_in[0];
  float* out = (float*)d_out;

  const size_t nBN = (size_t)NBATCH * NDET;
  char* w = (char*)d_ws;
  float4*       wsBox = (float4*)w;                 w += nBN * sizeof(float4);
  unsigned int* wsKey = (unsigned int*)w;           w += nBN * sizeof(unsigned int);
  int*          wsCls = (int*)w;                    w += nBN * sizeof(int);
  float4*       wsTB  = (float4*)w;                 w += (size_t)NBATCH * TOPK * sizeof(float4);
  float*        wsTS  = (float*)w;                  w += (size_t)NBATCH * TOPK * sizeof(float);
  int*          wsTC  = (int*)w;                    w += (size_t)NBATCH * TOPK * sizeof(int);
  int*          wsTV  = (int*)w;                    w += (size_t)NBATCH * TOPK * sizeof(int);

  // K1: 8 waves (rows) per 256-thread block
  int total = (int)nBN;
  int blocks1 = (total + 7) / 8;
  nms_decode<<<blocks1, 256, 0, stream>>>(x, wsKey, wsCls, wsBox, total);

  // K2: one block per batch, 192KB dynamic LDS (CDNA5: 320KB/WGP)
  size_t lds2 = (size_t)SORTM * 4 + (size_t)SORTM * 2;
  nms_topk<<<NBATCH, 1024, lds2, stream>>>(wsKey, wsCls, wsBox,
                                           wsTB, wsTS, wsTC, wsTV);

  // K3: one block per batch, boxes + suppression bitmask + keep words in LDS
  size_t lds3 = (size_t)TPAD * 16 + (size_t)TPAD * 32 * 4 + 32 * 4;
  nms_suppress<<<NBATCH, 1024, lds3, stream>>>(wsTB, wsTS, wsTC, wsTV, out);
}